// GNNConv1DEBMV2_15857019257195
// MI455X (gfx1250) — compile-verified
//
#include <hip/hip_runtime.h>
#include <math.h>

// CDNA5 / gfx1250. Wave32. FP32 WMMA (V_WMMA_F32_16X16X4_F32) for all GEMMs
// to match the fp32 reference exactly; VALU+v_exp for the SiLU/max combine.
// B-matrix weights are stored pre-swizzled so each WMMA B fragment is one
// aligned ds_load_b64 (pair {W[2k][col], W[2k+1][col]} adjacent in LDS).

typedef float v2f __attribute__((ext_vector_type(2)));
typedef float v8f __attribute__((ext_vector_type(8)));

#define BN   8
#define TT   16
#define NNN  64
#define CIN  64     // INP_DIM
#define COUT 32     // OUT_DIM
#define TDIM 32
#define FDIM 512
#define XROWS (BN*TT*NNN)      // 8192
#define PCROWS (BN*NNN*NNN)    // 32768

// swizzled offset of weight element (row c, col o) inside a 64x64 matrix:
// pairs of consecutive rows interleaved per column -> one b64 load per B frag.
__device__ __forceinline__ int wswz(int c, int o) {
  return ((c >> 1) * 128) + (o * 2) + (c & 1);
}

// ---------------- time embedding: te = gelu(emb@Wt1+bt1)@Wt2+bt2 ----------------
__global__ void k_te(const float* __restrict__ t, const float* __restrict__ Wt1,
                     const float* __restrict__ bt1, const float* __restrict__ Wt2,
                     const float* __restrict__ bt2, float* __restrict__ te) {
  __shared__ float emb[BN * FDIM];   // 16 KB
  __shared__ float h1[BN * TDIM];
  int tid = threadIdx.x;
  const float cst = -logf(10000.0f) / 255.0f;   // half-1 = 255
  for (int i = tid; i < BN * FDIM; i += 256) {
    int b = i >> 9, k = i & 511;
    float tv = t[b];
    if (k < 256) emb[i] = sinf(tv * expf((float)k * cst));
    else         emb[i] = cosf(tv * expf((float)(k - 256) * cst));
  }
  __syncthreads();
  {
    int b = tid >> 5, o = tid & 31;
    float acc = bt1[o];
    const float* e = emb + b * FDIM;
    for (int k = 0; k < FDIM; ++k) acc += e[k] * Wt1[k * TDIM + o];
    h1[b * TDIM + o] = 0.5f * acc * (1.0f + erff(acc * 0.70710678118f)); // exact GELU
  }
  __syncthreads();
  {
    int b = tid >> 5, o = tid & 31;
    float acc = bt2[o];
    const float* hh = h1 + b * TDIM;
    for (int k = 0; k < TDIM; ++k) acc += hh[k] * Wt2[k * TDIM + o];
    te[b * TDIM + o] = acc;
  }
}

// ------------- effective per-t weights: 2 layers x {p1,p2} x 3 variants -------------
// variant 0: t==0 (chunk A masked), 1: interior (A+B+C), 2: t==T-1 (chunk C masked)
// Output is written PRE-SWIZZLED (wswz layout) for direct b64 B-fragment loads.
__global__ void k_prep(const float* __restrict__ W1, const float* __restrict__ W2,
                       float* __restrict__ weff) {
  int idx = blockIdx.x * 256 + threadIdx.x;   // 12*4096
  if (idx >= 12 * 4096) return;
  int u = idx >> 12, e = idx & 4095;
  int c = e >> 6, o = e & 63;
  int l = u / 6, rem = u % 6;
  int which = rem / 3, v = rem % 3;           // which: 0=p1 (W[0:192]), 1=p2 (W[192:384])
  const float* W = (l == 0) ? W1 : W2;
  int off = which * 192;
  float acc = W[(off + 64 + c) * 64 + o];                   // middle chunk always
  if (v != 0) acc += W[(off + c) * 64 + o];                 // chunk A unless t==0
  if (v != 2) acc += W[(off + 128 + c) * 64 + o];           // chunk C unless t==T-1
  weff[u * 4096 + wswz(c, o)] = acc;
}

// ---------------- X0 = concat([out, broadcast(te)]) ----------------
__global__ void k_x0(const float* __restrict__ outp, const float* __restrict__ te,
                     float* __restrict__ X) {
  int idx = blockIdx.x * 256 + threadIdx.x;   // 8192*64
  int row = idx >> 6, c = idx & 63;
  int b = row >> 10;                          // T*N = 1024 rows per batch
  X[idx] = (c < COUT) ? outp[row * COUT + c] : te[b * TDIM + (c - COUT)];
}

// ---------------- P1/P2 GEMM: 16-row strips, t-uniform per block ----------------
// block = one (b,t): 64 rows, 4 waves x 16-row strips; swizzled weights in LDS.
__global__ void k_pgemm(const float* __restrict__ X, const float* __restrict__ weff_layer,
                        float* __restrict__ P1, float* __restrict__ P2) {
  __shared__ __align__(16) float w1s[4096];
  __shared__ __align__(16) float w2s[4096];
  int tid = threadIdx.x;
  int bid = blockIdx.x;                       // = b*T + t
  int t = bid & (TT - 1);
  int v = (t == 0) ? 0 : ((t == TT - 1) ? 2 : 1);
  const float* g1 = weff_layer + v * 4096;          // p1 variant (already swizzled)
  const float* g2 = weff_layer + (3 + v) * 4096;    // p2 variant
  for (int i = tid; i < 4096; i += 128) { w1s[i] = g1[i]; w2s[i] = g2[i]; }
  __syncthreads();

  int wave = tid >> 5, lane = tid & 31;
  int row0 = bid * 64 + wave * 16;
  int lr = lane & 15, lh = lane >> 4;

  // A fragments: 16x4 f32 tiles; lanes 0-15 hold K=k0,k0+1; lanes 16-31 hold k0+2,k0+3
  v2f a[16];
  const float* xrow = X + (row0 + lr) * 64 + lh * 2;
#pragma unroll
  for (int kk = 0; kk < 16; ++kk) a[kk] = *(const v2f*)(xrow + kk * 4);

  v8f acc1[4] = {}, acc2[4] = {};
#pragma unroll
  for (int kk = 0; kk < 16; ++kk) {
    int kpair = kk * 2 + lh;                  // = (kk*4 + lh*2) / 2
#pragma unroll
    for (int j = 0; j < 4; ++j) {
      int col = j * 16 + lr;
      v2f b1 = *(const v2f*)(w1s + kpair * 128 + col * 2);  // one ds_load_b64
      v2f b2 = *(const v2f*)(w2s + kpair * 128 + col * 2);
      acc1[j] = __builtin_amdgcn_wmma_f32_16x16x4_f32(false, a[kk], false, b1,
                                                      (short)0, acc1[j], false, false);
      acc2[j] = __builtin_amdgcn_wmma_f32_16x16x4_f32(false, a[kk], false, b2,
                                                      (short)0, acc2[j], false, false);
    }
  }
#pragma unroll
  for (int j = 0; j < 4; ++j) {
    int col = j * 16 + lr;
#pragma unroll
    for (int r = 0; r < 8; ++r) {
      int m = r + lh * 8;                    // C/D layout: VGPR r -> M=r (lanes 0-15), M=r+8 (16-31)
      P1[(row0 + m) * 64 + col] = acc1[j][r];
      P2[(row0 + m) * 64 + col] = acc2[j][r];
    }
  }
}

// ---------------- PC GEMM: inp@Wc1 + inp^T@Wc2 + bias ----------------
__global__ void k_pcgemm(const float* __restrict__ inp, const float* __restrict__ Wc1,
                         const float* __restrict__ Wc2, const float* __restrict__ bias,
                         float* __restrict__ PC) {
  __shared__ __align__(16) float w1s[4096];
  __shared__ __align__(16) float w2s[4096];
  __shared__ float bs[64];
  int tid = threadIdx.x;
  // stage with swizzle (one-time scatter)
  for (int i = tid; i < 4096; i += 128) {
    int c = i >> 6, o = i & 63;
    int d = wswz(c, o);
    w1s[d] = Wc1[i];
    w2s[d] = Wc2[i];
  }
  if (tid < 64) bs[tid] = bias[tid];
  __syncthreads();

  int wave = tid >> 5, lane = tid & 31;
  int s = blockIdx.x * 4 + wave;              // strip 0..2047
  int row0 = s * 16;                          // pc row = (b*64+m)*64+n
  int b = row0 >> 12;
  int m = (row0 >> 6) & 63;
  int n0 = row0 & 63;
  int lr = lane & 15, lh = lane >> 4;
  int n = n0 + lr;

  const float* a1p = inp + ((b * 64 + m) * 64 + n) * 64 + lh * 2;  // inp[b,m,n,:]
  const float* a2p = inp + ((b * 64 + n) * 64 + m) * 64 + lh * 2;  // inp[b,n,m,:]
  v2f a1[16], a2[16];
#pragma unroll
  for (int kk = 0; kk < 16; ++kk) {
    a1[kk] = *(const v2f*)(a1p + kk * 4);
    a2[kk] = *(const v2f*)(a2p + kk * 4);
  }
  v8f acc[4] = {};
#pragma unroll
  for (int kk = 0; kk < 16; ++kk) {
    int kpair = kk * 2 + lh;
#pragma unroll
    for (int j = 0; j < 4; ++j) {
      int col = j * 16 + lr;
      v2f b1 = *(const v2f*)(w1s + kpair * 128 + col * 2);
      v2f b2 = *(const v2f*)(w2s + kpair * 128 + col * 2);
      acc[j] = __builtin_amdgcn_wmma_f32_16x16x4_f32(false, a1[kk], false, b1,
                                                     (short)0, acc[j], false, false);
      acc[j] = __builtin_amdgcn_wmma_f32_16x16x4_f32(false, a2[kk], false, b2,
                                                     (short)0, acc[j], false, false);
    }
  }
#pragma unroll
  for (int j = 0; j < 4; ++j) {
    int col = j * 16 + lr;
#pragma unroll
    for (int r = 0; r < 8; ++r) {
      int mm = r + lh * 8;
      PC[(row0 + mm) * 64 + col] = acc[j][r] + bs[col];
    }
  }
}

// ---------------- combine: out[b,t,m,o] = max_n silu(p1+p2+pc) ----------------
// block = (b,m); pc tile (64x64) staged in LDS once; 4 t's per thread in regs.
__global__ void k_combine(const float* __restrict__ P1, const float* __restrict__ P2,
                          const float* __restrict__ PC, float* __restrict__ Xout) {
  __shared__ float pcs[4096];
  int tid = threadIdx.x;
  int bid = blockIdx.x;                        // b*64 + m
  int b = bid >> 6, m = bid & 63;
  const float* pcg = PC + bid * 4096;          // rows (b,m,*) contiguous
  for (int i = tid; i < 4096; i += 256) pcs[i] = pcg[i];
  __syncthreads();

  int o = tid & 63;
  int tg = tid >> 6;                           // 4 t-values per thread
  float p1v[4], acc[4];
#pragma unroll
  for (int i = 0; i < 4; ++i) {
    int t = tg * 4 + i;
    p1v[i] = P1[((b * TT + t) * NNN + m) * 64 + o];
    acc[i] = -3.4e38f;
  }
  for (int n = 0; n < 64; ++n) {
    float pcv = pcs[n * 64 + o];
#pragma unroll
    for (int i = 0; i < 4; ++i) {
      int t = tg * 4 + i;
      float h = p1v[i] + P2[((b * TT + t) * NNN + n) * 64 + o] + pcv;
      float sv = h / (1.0f + __expf(-h));      // SiLU via v_exp_f32
      acc[i] = fmaxf(acc[i], sv);
    }
  }
#pragma unroll
  for (int i = 0; i < 4; ++i) {
    int t = tg * 4 + i;
    Xout[((b * TT + t) * NNN + m) * 64 + o] = acc[i];
  }
}

// ---------------- final: out[b] = sum_{t,n} (x@Wfc) + T*N*bfc ----------------
__global__ void k_final(const float* __restrict__ X, const float* __restrict__ Wfc,
                        const float* __restrict__ bfc, float* __restrict__ outv) {
  __shared__ float red[256];
  __shared__ float wf[64];
  int tid = threadIdx.x, b = blockIdx.x;
  if (tid < 64) wf[tid] = Wfc[tid];
  __syncthreads();
  float s = 0.0f;
  for (int r = tid; r < TT * NNN; r += 256) {
    const float* xr = X + (b * TT * NNN + r) * 64;
    float d = 0.0f;
    for (int oc = 0; oc < 64; ++oc) d += xr[oc] * wf[oc];
    s += d;
  }
  red[tid] = s;
  __syncthreads();
  for (int st = 128; st > 0; st >>= 1) {
    if (tid < st) red[tid] += red[tid + st];
    __syncthreads();
  }
  if (tid == 0) outv[b] = red[0] + (float)(TT * NNN) * bfc[0];
}

extern "C" void kernel_launch(void* const* d_in, const int* in_sizes, int n_in,
                              void* d_out, int out_size, void* d_ws, size_t ws_size,
                              hipStream_t stream) {
  const float* inp  = (const float*)d_in[0];
  const float* outp = (const float*)d_in[1];
  const float* t    = (const float*)d_in[2];
  const float* Wt1  = (const float*)d_in[3];
  const float* bt1  = (const float*)d_in[4];
  const float* Wt2  = (const float*)d_in[5];
  const float* bt2  = (const float*)d_in[6];
  const float* W1   = (const float*)d_in[7];
  const float* b1   = (const float*)d_in[8];
  const float* W2   = (const float*)d_in[9];
  const float* b2   = (const float*)d_in[10];
  const float* Wfc  = (const float*)d_in[11];
  const float* bfc  = (const float*)d_in[12];
  float* ws = (float*)d_ws;

  float* te   = ws;                       // 256
  float* weff = ws + 256;                 // 12*4096 = 49152 (swizzled)
  float* Xa   = ws + 49408;               // 524288
  float* Xb   = Xa + 524288;              // 524288
  float* P1   = Xb + 524288;              // 524288
  float* P2   = P1 + 524288;              // 524288
  float* PC   = P2 + 524288;              // 2097152  (total ~17 MB)

  k_te  <<<1,    256, 0, stream>>>(t, Wt1, bt1, Wt2, bt2, te);
  k_prep<<<192,  256, 0, stream>>>(W1, W2, weff);
  k_x0  <<<2048, 256, 0, stream>>>(outp, te, Xa);
  // layer 1
  k_pgemm  <<<128, 128, 0, stream>>>(Xa, weff + 0 * 6 * 4096, P1, P2);
  k_pcgemm <<<512, 128, 0, stream>>>(inp, W1 + 384 * 64, W1 + 448 * 64, b1, PC);
  k_combine<<<512, 256, 0, stream>>>(P1, P2, PC, Xb);
  // layer 2
  k_pgemm  <<<128, 128, 0, stream>>>(Xb, weff + 1 * 6 * 4096, P1, P2);
  k_pcgemm <<<512, 128, 0, stream>>>(inp, W2 + 384 * 64, W2 + 448 * 64, b2, PC);
  k_combine<<<512, 256, 0, stream>>>(P1, P2, PC, Xa);
  k_final  <<<8,   256, 0, stream>>>(Xa, Wfc, bfc, (float*)d_out);
}